// CNNPolicy_45784351375529
// MI455X (gfx1250) — compile-verified
//
#include <hip/hip_runtime.h>
#include <math.h>

typedef float v2f __attribute__((ext_vector_type(2)));
typedef float v8f __attribute__((ext_vector_type(8)));

#define LOG2PI_F 1.8378770664093453f

// ---------------- LDS layout (floats) ----------------
#define WAVES_PER_WG 4
#define LDS_W1 0                         // 256 x 48 (k padded 46->48 with zeros)
#define LDS_W2 (LDS_W1 + 256 * 48)      // 128 x 260
#define LDS_W3 (LDS_W2 + 128 * 260)     // 64 x 128 (or 1 x 128 for critic)
#define LDS_B1 (LDS_W3 + 64 * 128)      // 256
#define LDS_B2 (LDS_B1 + 256)           // 128
#define LDS_B3 (LDS_B2 + 128)           // 64
#define LDS_H1 (LDS_B3 + 64)            // per wave: 16 x 260
#define LDS_H2 (LDS_H1 + WAVES_PER_WG * 16 * 260)  // per wave: 16 x 128
#define LDS_TOTAL (LDS_H2 + WAVES_PER_WG * 16 * 128)
// LDS_TOTAL = 79040 floats = 316160 bytes  (<= 320KB WGP LDS)

enum { MODE_MSG_SELF = 0, MODE_MSG_NEI = 1, MODE_ACT = 2, MODE_CRITIC = 3 };

static __device__ __forceinline__ v8f wmma_k4(v2f a, v2f b, v8f c) {
  // D(16x16) = A(16x4) * B(4x16) + C ; fp32 WMMA, full-rate vs memory-bound problem
  return __builtin_amdgcn_wmma_f32_16x16x4_f32(false, a, false, b, (short)0, c,
                                               false, false);
}

template <int MODE>
__global__ __launch_bounds__(128) void trunk_kernel(
    const float* __restrict__ obs,     // rows x 46
    int rows,
    const float* __restrict__ goal,    // B x 2
    const float* __restrict__ speed,   // B x 2
    const float* __restrict__ w1, const float* __restrict__ b1,   // 256x46, 256
    const float* __restrict__ w2, const float* __restrict__ b2,   // 128x260, 128
    const float* __restrict__ w3, const float* __restrict__ b3,   // 64x128 / 1x128
    const float* __restrict__ logstd,                            // 64 (msg modes)
    const float* __restrict__ eps,                               // rows x 64
    const unsigned char* __restrict__ mask,                      // B x 16 (nei)
    float* __restrict__ out) {
  extern __shared__ float smem[];
  const int tid = threadIdx.x;

  // ---- cooperative one-time weight staging into LDS ----
  for (int i = tid; i < 256 * 48; i += 128) {
    int n = i / 48, k = i - n * 48;
    smem[LDS_W1 + i] = (k < 46) ? w1[(size_t)n * 46 + k] : 0.0f;
  }
  for (int i = tid; i < 128 * 260; i += 128) smem[LDS_W2 + i] = w2[i];
  if (MODE == MODE_CRITIC) {
    for (int i = tid; i < 128; i += 128) smem[LDS_W3 + i] = w3[i];
    if (tid == 0) smem[LDS_B3] = b3[0];
  } else {
    for (int i = tid; i < 64 * 128; i += 128) smem[LDS_W3 + i] = w3[i];
    for (int i = tid; i < 64; i += 128) smem[LDS_B3 + i] = b3[i];
  }
  for (int i = tid; i < 256; i += 128) smem[LDS_B1 + i] = b1[i];
  for (int i = tid; i < 128; i += 128) smem[LDS_B2 + i] = b2[i];
  __syncthreads();

  const int lane = tid & 31;
  const int wv   = tid >> 5;
  const int l16  = lane & 15;
  const int half = lane >> 4;   // 0: K lanes {0,1}, 1: K lanes {2,3}
  const int off0 = half * 2;

  float* h1 = &smem[LDS_H1 + wv * 16 * 260];  // 16 rows x 260 (h1 + goal/speed)
  float* h2 = &smem[LDS_H2 + wv * 16 * 128];  // 16 rows x 128

  const int ntiles = rows >> 4;
  for (int t = blockIdx.x * WAVES_PER_WG + wv; t < ntiles;
       t += gridDim.x * WAVES_PER_WG) {
    const int row0 = t * 16;

    // ---- preload A fragments for layer1 (16x48, padded) ----
    float a1[24];
    {
      const float* op = obs + (size_t)(row0 + l16) * 46;
      #pragma unroll
      for (int s = 0; s < 12; ++s) {
        int k = 4 * s + off0;
        a1[2 * s]     = (k     < 46) ? op[k]     : 0.0f;
        a1[2 * s + 1] = (k + 1 < 46) ? op[k + 1] : 0.0f;
      }
    }

    // ---- layer1: 46 -> 256, relu ----
    #pragma unroll 1
    for (int nt = 0; nt < 16; ++nt) {
      const int n = nt * 16 + l16;
      const float bias = smem[LDS_B1 + n];
      v8f c;
      #pragma unroll
      for (int j = 0; j < 8; ++j) c[j] = bias;
      const float* wr = &smem[LDS_W1 + n * 48 + off0];
      #pragma unroll
      for (int s = 0; s < 12; ++s) {
        v2f a; a[0] = a1[2 * s]; a[1] = a1[2 * s + 1];
        v2f b; b[0] = wr[4 * s]; b[1] = wr[4 * s + 1];
        c = wmma_k4(a, b, c);
      }
      #pragma unroll
      for (int j = 0; j < 8; ++j) {
        int m = j + half * 8;
        float v = c[j];
        h1[m * 260 + n] = v > 0.0f ? v : 0.0f;
      }
    }
    // concat goal,speed as columns 256..259
    if (lane < 16) {
      int gr = row0 + lane;
      int bb = (MODE == MODE_MSG_NEI) ? (gr >> 4) : gr;
      h1[lane * 260 + 256] = goal[(size_t)bb * 2 + 0];
      h1[lane * 260 + 257] = goal[(size_t)bb * 2 + 1];
      h1[lane * 260 + 258] = speed[(size_t)bb * 2 + 0];
      h1[lane * 260 + 259] = speed[(size_t)bb * 2 + 1];
    }

    // ---- layer2: 260 -> 128, relu (s-outer, 8 accumulators, A loaded once) ----
    v8f c2[8];
    #pragma unroll
    for (int nt = 0; nt < 8; ++nt) {
      const float bias = smem[LDS_B2 + nt * 16 + l16];
      #pragma unroll
      for (int j = 0; j < 8; ++j) c2[nt][j] = bias;
    }
    #pragma unroll 1
    for (int s = 0; s < 65; ++s) {
      int k = 4 * s + off0;
      v2f a; a[0] = h1[l16 * 260 + k]; a[1] = h1[l16 * 260 + k + 1];
      #pragma unroll
      for (int nt = 0; nt < 8; ++nt) {
        const float* wr = &smem[LDS_W2 + (nt * 16 + l16) * 260 + k];
        v2f b; b[0] = wr[0]; b[1] = wr[1];
        c2[nt] = wmma_k4(a, b, c2[nt]);
      }
    }
    #pragma unroll
    for (int nt = 0; nt < 8; ++nt) {
      #pragma unroll
      for (int j = 0; j < 8; ++j) {
        int m = j + half * 8;
        float v = c2[nt][j];
        h2[m * 128 + nt * 16 + l16] = v > 0.0f ? v : 0.0f;
      }
    }

    if (MODE == MODE_CRITIC) {
      // ---- critic head: v = h2 . c_w3 + c_b3 (128 -> 1) ----
      float sum = 0.0f;
      const int kbase = half * 64;
      #pragma unroll 4
      for (int k = 0; k < 64; ++k)
        sum += h2[l16 * 128 + kbase + k] * smem[LDS_W3 + kbase + k];
      sum += __shfl_xor(sum, 16, 32);
      if (lane < 16) out[row0 + lane] = sum + smem[LDS_B3];
    } else {
      // ---- layer3: 128 -> 64 (s-outer, 4 accumulators) ----
      v8f c3[4];
      #pragma unroll
      for (int nt = 0; nt < 4; ++nt) {
        const float bias = smem[LDS_B3 + nt * 16 + l16];
        #pragma unroll
        for (int j = 0; j < 8; ++j) c3[nt][j] = bias;
      }
      #pragma unroll 1
      for (int s = 0; s < 32; ++s) {
        int k = 4 * s + off0;
        v2f a; a[0] = h2[l16 * 128 + k]; a[1] = h2[l16 * 128 + k + 1];
        #pragma unroll
        for (int nt = 0; nt < 4; ++nt) {
          const float* wr = &smem[LDS_W3 + (nt * 16 + l16) * 128 + k];
          v2f b; b[0] = wr[0]; b[1] = wr[1];
          c3[nt] = wmma_k4(a, b, c3[nt]);
        }
      }
      #pragma unroll
      for (int nt = 0; nt < 4; ++nt) {
        const int n = nt * 16 + l16;
        if (MODE == MODE_ACT) {
          #pragma unroll
          for (int j = 0; j < 8; ++j) {
            int gr = row0 + j + half * 8;
            float v = c3[nt][j];
            out[(size_t)gr * 64 + n] = v > 0.0f ? v : 0.0f;
          }
        } else if (MODE == MODE_MSG_SELF) {
          const float sdev = expf(logstd[n]);
          #pragma unroll
          for (int j = 0; j < 8; ++j) {
            int gr = row0 + j + half * 8;
            out[(size_t)gr * 64 + n] =
                tanhf(c3[nt][j]) + sdev * eps[(size_t)gr * 64 + n];
          }
        } else {  // MODE_MSG_NEI: tile == one batch's 16 neighbors; masked sum
          const float sdev = expf(logstd[n]);
          const int bb = t;  // batch index
          float colsum = 0.0f;
          #pragma unroll
          for (int j = 0; j < 8; ++j) {
            int m = j + half * 8;
            int gr = row0 + m;
            float d = tanhf(c3[nt][j]) + sdev * eps[(size_t)gr * 64 + n];
            colsum += mask[(size_t)bb * 16 + m] ? d : 0.0f;
          }
          colsum += __shfl_xor(colsum, 16, 32);
          if (lane < 16) out[(size_t)bb * 64 + nt * 16 + lane] = colsum;
        }
      }
    }
  }
}

// ---------------- fusion head: per-batch tiny MLP ----------------
__global__ __launch_bounds__(128) void head_kernel(
    const float* __restrict__ a_act,     // B x 64
    const float* __restrict__ self_msg,  // B x 64
    const float* __restrict__ nsum,      // B x 64
    const float* __restrict__ goal, const float* __restrict__ speed,
    const float* __restrict__ f_w1, const float* __restrict__ f_b1,  // 64x192
    const float* __restrict__ f_w2, const float* __restrict__ f_b2,  // 128x68
    const float* __restrict__ mu_w, const float* __restrict__ mu_b,  // 2x128
    const float* __restrict__ ca_logstd, const float* __restrict__ p_logstd,
    const float* __restrict__ eps_act,   // B x 2
    float* __restrict__ action_out, float* __restrict__ logprob_out,
    float* __restrict__ mean_out) {
  __shared__ float xx[192];
  __shared__ float act68[68];
  __shared__ float act2[128];
  __shared__ float red[2];
  const int b = blockIdx.x;
  const int tid = threadIdx.x;

  if (tid < 64) {
    xx[tid]       = a_act[(size_t)b * 64 + tid];
    xx[64 + tid]  = self_msg[(size_t)b * 64 + tid];
    xx[128 + tid] = nsum[(size_t)b * 64 + tid];
  }
  __syncthreads();
  if (tid < 64) {
    float s = f_b1[tid];
    const float* wr = f_w1 + (size_t)tid * 192;
    #pragma unroll 4
    for (int k = 0; k < 192; ++k) s += wr[k] * xx[k];
    act68[tid] = tanhf(s);
  } else if (tid < 68) {
    int c = tid - 64;
    float g = (c < 2) ? goal[(size_t)b * 2 + c] : speed[(size_t)b * 2 + c - 2];
    act68[tid] = tanhf(g);
  }
  __syncthreads();
  {
    float s = f_b2[tid];
    const float* wr = f_w2 + (size_t)tid * 68;
    #pragma unroll 4
    for (int k = 0; k < 68; ++k) s += wr[k] * act68[k];
    act2[tid] = tanhf(s);
  }
  __syncthreads();
  if (tid < 2) {
    float s = mu_b[tid];
    const float* wr = mu_w + (size_t)tid * 128;
    #pragma unroll 4
    for (int k = 0; k < 128; ++k) s += wr[k] * act2[k];
    const float mean = s;
    const float diff = expf(ca_logstd[tid]) * eps_act[(size_t)b * 2 + tid];
    mean_out[(size_t)b * 2 + tid]   = mean;
    action_out[(size_t)b * 2 + tid] = mean + diff;
    const float lsd = p_logstd[tid];
    const float var = expf(2.0f * lsd);
    red[tid] = -(diff * diff) / (2.0f * var) - 0.5f * LOG2PI_F - lsd;
  }
  __syncthreads();
  if (tid == 0) logprob_out[b] = red[0] + red[1];
}

extern "C" void kernel_launch(void* const* d_in, const int* in_sizes, int n_in,
                              void* d_out, int out_size, void* d_ws,
                              size_t ws_size, hipStream_t stream) {
  const int B = in_sizes[1] / 2;  // goal is B x 2
  const int K = 16;

  const float* x        = (const float*)d_in[0];
  const float* goal     = (const float*)d_in[1];
  const float* speed    = (const float*)d_in[2];
  const float* nobs     = (const float*)d_in[3];
  const float* eps_msg  = (const float*)d_in[4];
  const float* eps_nei  = (const float*)d_in[5];
  const float* eps_act  = (const float*)d_in[6];
  const float* m_w1 = (const float*)d_in[7];
  const float* m_b1 = (const float*)d_in[8];
  const float* m_w2 = (const float*)d_in[9];
  const float* m_b2 = (const float*)d_in[10];
  const float* m_w3 = (const float*)d_in[11];
  const float* m_b3 = (const float*)d_in[12];
  const float* m_logstd = (const float*)d_in[13];
  const float* a_w1 = (const float*)d_in[14];
  const float* a_b1 = (const float*)d_in[15];
  const float* a_w2 = (const float*)d_in[16];
  const float* a_b2 = (const float*)d_in[17];
  const float* a_w3 = (const float*)d_in[18];
  const float* a_b3 = (const float*)d_in[19];
  const float* f_w1 = (const float*)d_in[20];
  const float* f_b1 = (const float*)d_in[21];
  const float* f_w2 = (const float*)d_in[22];
  const float* f_b2 = (const float*)d_in[23];
  const float* mu_w = (const float*)d_in[24];
  const float* mu_b = (const float*)d_in[25];
  const float* ca_logstd = (const float*)d_in[26];
  const float* p_logstd  = (const float*)d_in[27];
  const float* c_w1 = (const float*)d_in[28];
  const float* c_b1 = (const float*)d_in[29];
  const float* c_w2 = (const float*)d_in[30];
  const float* c_b2 = (const float*)d_in[31];
  const float* c_w3 = (const float*)d_in[32];
  const float* c_b3 = (const float*)d_in[33];
  const unsigned char* mask = (const unsigned char*)d_in[34];

  float* ws = (float*)d_ws;
  float* self_msg = ws;                        // B*64
  float* nsum     = ws + (size_t)B * 64;       // B*64
  float* a_act    = ws + (size_t)2 * B * 64;   // B*64

  float* outp        = (float*)d_out;
  float* v_out       = outp;               // B
  float* action_out  = outp + (size_t)B;   // 2B
  float* logprob_out = outp + (size_t)3 * B;  // B
  float* mean_out    = outp + (size_t)4 * B;  // 2B

  const size_t smem = (size_t)LDS_TOTAL * sizeof(float);
  const int tilesS = B / 16;
  const int tilesN = (B * K) / 16;
  int gS = (tilesS + WAVES_PER_WG - 1) / WAVES_PER_WG;
  if (gS > 2048) gS = 2048;
  int gN = (tilesN + WAVES_PER_WG - 1) / WAVES_PER_WG;
  if (gN > 2048) gN = 2048;

  trunk_kernel<MODE_MSG_SELF><<<gS, 128, smem, stream>>>(
      x, B, goal, speed, m_w1, m_b1, m_w2, m_b2, m_w3, m_b3, m_logstd, eps_msg,
      nullptr, self_msg);
  trunk_kernel<MODE_MSG_NEI><<<gN, 128, smem, stream>>>(
      nobs, B * K, goal, speed, m_w1, m_b1, m_w2, m_b2, m_w3, m_b3, m_logstd,
      eps_nei, mask, nsum);
  trunk_kernel<MODE_ACT><<<gS, 128, smem, stream>>>(
      x, B, goal, speed, a_w1, a_b1, a_w2, a_b2, a_w3, a_b3, nullptr, nullptr,
      nullptr, a_act);
  trunk_kernel<MODE_CRITIC><<<gS, 128, smem, stream>>>(
      x, B, goal, speed, c_w1, c_b1, c_w2, c_b2, c_w3, c_b3, nullptr, nullptr,
      nullptr, v_out);

  head_kernel<<<B, 128, 0, stream>>>(a_act, self_msg, nsum, goal, speed, f_w1,
                                     f_b1, f_w2, f_b2, mu_w, mu_b, ca_logstd,
                                     p_logstd, eps_act, action_out, logprob_out,
                                     mean_out);
}